// BendingModel_30167850287109
// MI455X (gfx1250) — compile-verified
//
#include <hip/hip_runtime.h>

#define Bn   8
#define Sn   32
#define Nn   512
#define FIN  24
#define Cc   56
#define EBn  262144
#define ESn  8192
#define NODES (Bn * Sn * Nn)       /* 131072 total nodes */
#define NPG_BEND (Sn * Nn)         /* 16384 nodes per bend graph */

typedef float v2f __attribute__((ext_vector_type(2)));
typedef float v8f __attribute__((ext_vector_type(8)));

// ---------- monotone float<->uint encoding for atomicMax on floats ----------
__device__ __forceinline__ unsigned fenc(float f) {
  unsigned u = __float_as_uint(f);
  return (u & 0x80000000u) ? ~u : (u | 0x80000000u);
}
__device__ __forceinline__ float fdec(unsigned u) {
  unsigned b = (u & 0x80000000u) ? (u & 0x7fffffffu) : ~u;
  return __uint_as_float(b);
}

// ---------- tiny prep: collapse edge path to per-edge 4-dot; softmax(mix) ---
// consts: [0..3]=v_b [4]=c_b [5..8]=v_s [9]=c_s [10]=w0 [11]=w1
__global__ void prep_kernel(const float* __restrict__ enc_W,
                            const float* __restrict__ enc_b,
                            const float* __restrict__ We_b,
                            const float* __restrict__ a_e_b,
                            const float* __restrict__ We_s,
                            const float* __restrict__ a_e_s,
                            const float* __restrict__ mix_w,
                            float* __restrict__ consts) {
  if (threadIdx.x == 0 && blockIdx.x == 0) {
    float WeA[4];
    // bend: WeA = We_b @ a_e_b  (We_b is [4][56] row-major)
    for (int k = 0; k < 4; ++k) {
      float s = 0.f;
      for (int c = 0; c < Cc; ++c) s += We_b[k * Cc + c] * a_e_b[c];
      WeA[k] = s;
    }
    for (int j = 0; j < 4; ++j) {
      float s = 0.f;
      for (int k = 0; k < 4; ++k) s += enc_W[j * 4 + k] * WeA[k];
      consts[j] = s;
    }
    {
      float s = 0.f;
      for (int k = 0; k < 4; ++k) s += enc_b[k] * WeA[k];
      consts[4] = s;
    }
    // section
    for (int k = 0; k < 4; ++k) {
      float s = 0.f;
      for (int c = 0; c < Cc; ++c) s += We_s[k * Cc + c] * a_e_s[c];
      WeA[k] = s;
    }
    for (int j = 0; j < 4; ++j) {
      float s = 0.f;
      for (int k = 0; k < 4; ++k) s += enc_W[j * 4 + k] * WeA[k];
      consts[5 + j] = s;
    }
    {
      float s = 0.f;
      for (int k = 0; k < 4; ++k) s += enc_b[k] * WeA[k];
      consts[9] = s;
    }
    float m  = fmaxf(mix_w[0], mix_w[1]);
    float e0 = expf(mix_w[0] - m), e1 = expf(mix_w[1] - m);
    consts[10] = e0 / (e0 + e1);
    consts[11] = e1 / (e0 + e1);
  }
}

// ---------- node transform h = x @ W via V_WMMA_F32_16X16X4_F32 ------------
// One wave per 16-row tile; 4 N-tiles of 16 (56 zero-padded to 64 in LDS);
// 6 K-steps of 4. W staged once per block into LDS, pair-interleaved so each
// B-fragment is a single unconditional ds_load_b64.
__global__ __launch_bounds__(256) void node_transform_wmma(
    const float* __restrict__ x, const float* __restrict__ W,
    float* __restrict__ h) {
  // LDS: 12 K-pairs x 64 cols x 2 floats = 1536 floats (6 KB)
  // lds[p*128 + col*2 + j] = W[2p + j][col]  (zero for col >= 56)
  __shared__ float Wlds[12 * 64 * 2];
  for (int i = threadIdx.x; i < 12 * 64 * 2; i += 256) {
    int j   = i & 1;
    int col = (i >> 1) & 63;
    int k   = ((i >> 7) << 1) + j;
    Wlds[i] = (col < Cc) ? W[k * Cc + col] : 0.0f;
  }
  __syncthreads();

  const int wave = (blockIdx.x * blockDim.x + threadIdx.x) >> 5;
  const int lane = threadIdx.x & 31;
  if (wave >= NODES / 16) return;
  const int mrow = lane & 15;      // N (and M for A-frag) index within tile
  const int half = lane >> 4;      // which K-pair of the 4-wide K step

  v8f acc[4] = {};
  const float* xrow = x + (wave * 16 + mrow) * FIN;

#pragma unroll
  for (int kk = 0; kk < 6; ++kk) {
    const int k0 = kk * 4 + half * 2;   // K base for this lane's pair
    const int p  = kk * 2 + half;       // K-pair index into LDS
    v2f a;
    a.x = xrow[k0];       // A[M=mrow][K=k0]
    a.y = xrow[k0 + 1];   // A[M=mrow][K=k0+1]
#pragma unroll
    for (int nt = 0; nt < 4; ++nt) {
      const int col = nt * 16 + mrow;
      v2f b = *(const v2f*)(Wlds + p * 128 + col * 2);  // ds_load_b64
      acc[nt] = __builtin_amdgcn_wmma_f32_16x16x4_f32(
          false, a, false, b, (short)0, acc[nt], false, false);
    }
  }

  // C/D layout: VGPR r -> M = r + 8*(lane/16), N = lane%16
  const int mbase = half << 3;
#pragma unroll
  for (int nt = 0; nt < 4; ++nt) {
    const int col = nt * 16 + mrow;
    if (col < Cc) {
#pragma unroll
      for (int r = 0; r < 8; ++r) {
        h[(wave * 16 + mbase + r) * Cc + col] = acc[nt][r];
      }
    }
  }
}

// ---------- per-node attention projections ---------------------------------
__global__ void alphas_kernel(const float* __restrict__ h,
                              const float* __restrict__ a_src,
                              const float* __restrict__ a_dst,
                              float* __restrict__ asrc,
                              float* __restrict__ adst) {
  int n = blockIdx.x * blockDim.x + threadIdx.x;
  if (n >= NODES) return;
  const float* hp = h + n * Cc;
  float s1 = 0.f, s2 = 0.f;
  for (int c = 0; c < Cc; ++c) {
    float v = hp[c];
    s1 += v * a_src[c];
    s2 += v * a_dst[c];
  }
  asrc[n] = s1;
  adst[n] = s2;
}

// ---------- init helpers ----------------------------------------------------
__global__ void init_maxsum(unsigned* __restrict__ maxb,
                            float* __restrict__ sumb, int n) {
  int i = blockIdx.x * blockDim.x + threadIdx.x;
  if (i < n) { maxb[i] = 0u; sumb[i] = 0.f; }
}

__global__ void init_acc(float* __restrict__ acc,
                         const float* __restrict__ bias, int nodes) {
  int i = blockIdx.x * blockDim.x + threadIdx.x;
  if (i < nodes * Cc) acc[i] = bias[i % Cc];
}

// ---------- edge pass 1: logits + segment max ------------------------------
__global__ void edge_logits(const int* __restrict__ eidx,
                            const float* __restrict__ attr,
                            const float* __restrict__ asrc,
                            const float* __restrict__ adst,
                            const float* __restrict__ consts, int cbase,
                            float* __restrict__ ebuf,
                            unsigned* __restrict__ maxb,
                            int E, int num_graphs, int attr_div, int npg) {
  int tid = blockIdx.x * blockDim.x + threadIdx.x;
  int total = E * num_graphs;
  if (tid >= total) return;
  int g = tid / E, e = tid - g * E;
  int src = eidx[e], dst = eidx[E + e];
  int ab = g / attr_div;
  const float* ap = attr + (ab * E + e) * 4;
  float z = asrc[g * npg + src] + adst[g * npg + dst] +
            ap[0] * consts[cbase] + ap[1] * consts[cbase + 1] +
            ap[2] * consts[cbase + 2] + ap[3] * consts[cbase + 3] +
            consts[cbase + 4];
  z = (z > 0.f) ? z : 0.2f * z;  // leaky_relu(0.2)
  ebuf[tid] = z;
  atomicMax(maxb + g * npg + dst, fenc(z));
}

// ---------- edge pass 2: exp + segment sum ---------------------------------
__global__ void edge_exp(const int* __restrict__ eidx,
                         const unsigned* __restrict__ maxb,
                         float* __restrict__ ebuf,
                         float* __restrict__ sumb,
                         int E, int num_graphs, int npg) {
  int tid = blockIdx.x * blockDim.x + threadIdx.x;
  int total = E * num_graphs;
  if (tid >= total) return;
  int g = tid / E, e = tid - g * E;
  int dst = eidx[E + e];
  float m  = fdec(maxb[g * npg + dst]);
  float ex = expf(ebuf[tid] - m);
  ebuf[tid] = ex;
  atomicAdd(sumb + g * npg + dst, ex);
}

// ---------- edge pass 3: weighted scatter-add of h[src] --------------------
// 4 threads per edge, 14 channels each (56 total).
__global__ void edge_accum(const int* __restrict__ eidx,
                           const float* __restrict__ ebuf,
                           const float* __restrict__ sumb,
                           const float* __restrict__ h,
                           float* __restrict__ acc,
                           int E, int num_graphs, int npg) {
  int tid = blockIdx.x * blockDim.x + threadIdx.x;
  int total = E * num_graphs * 4;
  if (tid >= total) return;
  int et = tid >> 2, cg = tid & 3;
  int g = et / E, e = et - g * E;
  int src = eidx[e], dst = eidx[E + e];
  float att = ebuf[et] / (sumb[g * npg + dst] + 1e-16f);
  const float* hp = h + (g * npg + src) * Cc;
  float* op = acc + (g * npg + dst) * Cc;
  int c0 = cg * 14;
#pragma unroll
  for (int c = 0; c < 14; ++c) {
    atomicAdd(op + c0 + c, att * hp[c0 + c]);
  }
}

// ---------- final mix: out = w0*leaky(bend,0.01) + w1*section --------------
__global__ void mix_kernel(const float* __restrict__ outb,
                           float* __restrict__ dout,
                           const float* __restrict__ consts) {
  int i = blockIdx.x * blockDim.x + threadIdx.x;
  if (i >= NODES * Cc) return;
  float w0 = consts[10], w1 = consts[11];
  float xb = outb[i];
  xb = (xb > 0.f) ? xb : 0.01f * xb;
  dout[i] = w0 * xb + w1 * dout[i];
}

// ===========================================================================
extern "C" void kernel_launch(void* const* d_in, const int* in_sizes, int n_in,
                              void* d_out, int out_size, void* d_ws, size_t ws_size,
                              hipStream_t stream) {
  const float* x       = (const float*)d_in[0];
  const int*   bei     = (const int*)d_in[1];
  const int*   sei     = (const int*)d_in[2];
  const float* beattr  = (const float*)d_in[3];
  const float* seattr  = (const float*)d_in[4];
  const float* enc_W   = (const float*)d_in[5];
  const float* enc_b   = (const float*)d_in[6];
  const float* Wb      = (const float*)d_in[7];
  const float* a_src_b = (const float*)d_in[8];
  const float* a_dst_b = (const float*)d_in[9];
  const float* We_b    = (const float*)d_in[10];
  const float* a_e_b   = (const float*)d_in[11];
  const float* bias_b  = (const float*)d_in[12];
  const float* Ws      = (const float*)d_in[13];
  const float* a_src_s = (const float*)d_in[14];
  const float* a_dst_s = (const float*)d_in[15];
  const float* We_s    = (const float*)d_in[16];
  const float* a_e_s   = (const float*)d_in[17];
  const float* bias_s  = (const float*)d_in[18];
  const float* mix_w   = (const float*)d_in[19];
  float* out = (float*)d_out;

  // workspace layout (floats); ~69.2 MB total, L2-resident on MI455X
  float*    wsf    = (float*)d_ws;
  float*    consts = wsf;                      // 16
  float*    bufA   = wsf + 16;                 // h (current graph)   7,340,032
  float*    bufB   = bufA + NODES * Cc;        // bend accumulator    7,340,032
  float*    asrc   = bufB + NODES * Cc;        // 131072
  float*    adst   = asrc + NODES;             // 131072
  unsigned* maxb   = (unsigned*)(adst + NODES);// 131072
  float*    sumb   = (float*)(maxb + NODES);   // 131072
  float*    ebuf   = sumb + NODES;             // 2,097,152

  const int TPB = 256;
  const int waves_blocks = (NODES / 16) / 8;             // 1024 (8 waves/block)
  const int node_blocks  = NODES / TPB;                  // 512
  const int elem_blocks  = (NODES * Cc + TPB - 1) / TPB; // 28672
  const int totS = ESn * Bn * Sn;                        // 2,097,152
  const int totB = EBn * Bn;                             // 2,097,152

  prep_kernel<<<1, 64, 0, stream>>>(enc_W, enc_b, We_b, a_e_b, We_s, a_e_s,
                                    mix_w, consts);

  // ---------------- section phase (accumulates into d_out) ----------------
  node_transform_wmma<<<waves_blocks, TPB, 0, stream>>>(x, Ws, bufA);
  alphas_kernel<<<node_blocks, TPB, 0, stream>>>(bufA, a_src_s, a_dst_s, asrc, adst);
  init_maxsum<<<node_blocks, TPB, 0, stream>>>(maxb, sumb, NODES);
  init_acc<<<elem_blocks, TPB, 0, stream>>>(out, bias_s, NODES);
  edge_logits<<<totS / TPB, TPB, 0, stream>>>(sei, seattr, asrc, adst, consts, 5,
                                              ebuf, maxb, ESn, Bn * Sn, Sn, Nn);
  edge_exp<<<totS / TPB, TPB, 0, stream>>>(sei, maxb, ebuf, sumb, ESn, Bn * Sn, Nn);
  edge_accum<<<(totS * 4) / TPB, TPB, 0, stream>>>(sei, ebuf, sumb, bufA, out,
                                                   ESn, Bn * Sn, Nn);

  // ---------------- bend phase (accumulates into bufB) --------------------
  node_transform_wmma<<<waves_blocks, TPB, 0, stream>>>(x, Wb, bufA);
  alphas_kernel<<<node_blocks, TPB, 0, stream>>>(bufA, a_src_b, a_dst_b, asrc, adst);
  init_maxsum<<<node_blocks, TPB, 0, stream>>>(maxb, sumb, NODES);
  init_acc<<<elem_blocks, TPB, 0, stream>>>(bufB, bias_b, NODES);
  edge_logits<<<totB / TPB, TPB, 0, stream>>>(bei, beattr, asrc, adst, consts, 0,
                                              ebuf, maxb, EBn, Bn, 1, NPG_BEND);
  edge_exp<<<totB / TPB, TPB, 0, stream>>>(bei, maxb, ebuf, sumb, EBn, Bn, NPG_BEND);
  edge_accum<<<(totB * 4) / TPB, TPB, 0, stream>>>(bei, ebuf, sumb, bufA, bufB,
                                                   EBn, Bn, NPG_BEND);

  // ---------------- mix ----------------------------------------------------
  mix_kernel<<<elem_blocks, TPB, 0, stream>>>(bufB, out, consts);
}